// MonocularBevModel_23905787969790
// MI455X (gfx1250) — compile-verified
//
#include <hip/hip_runtime.h>
#include <cstdint>

// Problem constants (from reference)
#define BB      4
#define DD      48
#define HH      32
#define WWID    88
#define CC      64
#define NXG     200
#define NYG     200
#define HW      (HH * WWID)          // 2816
#define DHW     (DD * HW)            // 135168
#define NPRIME  (BB * DHW)           // 540672
#define PLANE   (NXG * NYG)          // 40000
#define GRID_ELEMS (BB * PLANE * CC) // 10,240,000 floats (40.96 MB)

typedef float f2 __attribute__((ext_vector_type(2)));
typedef float f4 __attribute__((ext_vector_type(4)));

// ---------------------------------------------------------------------------
// Zero-fill (b128 stores, fully coalesced)
// ---------------------------------------------------------------------------
__global__ void bev_zero_f4(f4* __restrict__ p, int n4) {
    int i = blockIdx.x * blockDim.x + threadIdx.x;
    if (i < n4) p[i] = (f4){0.f, 0.f, 0.f, 0.f};
}

// ---------------------------------------------------------------------------
// Native f32 atomic add (global_atomic_add_f32, resolves in L2 atomic units)
// ---------------------------------------------------------------------------
__device__ __forceinline__ void fadd_native(float* p, float v) {
    unsafeAtomicAdd(p, v);
}

// ---------------------------------------------------------------------------
// Scatter-add: one wave32 per point; lane l handles channels 2l, 2l+1.
// Per-point scalars are forced onto the SALU/SMEM path via readfirstlane:
// index math, mask load, geom loads and the keep-test are wave-scalar, so
// rejected points cost one scalar load + scalar branch, no VMEM.
// Accumulator layout grid[b][gx][gy][c]: a point's 64 atomics land on 4
// consecutive cachelines (maximal L2 atomic coalescing).
// ---------------------------------------------------------------------------
__global__ void bev_scatter_vox(const float* __restrict__ x,
                                const float* __restrict__ geom,
                                const float* __restrict__ sem,
                                const float* __restrict__ dx,
                                const float* __restrict__ bx,
                                float* __restrict__ grid) {
    unsigned tid  = blockIdx.x * blockDim.x + threadIdx.x;
    unsigned lane = tid & 31u;
    // p is identical across the wave; make it provably scalar.
    unsigned p = (unsigned)__builtin_amdgcn_readfirstlane((int)(tid >> 5));
    if (p >= NPRIME) return;

    unsigned b  = p / DHW;
    unsigned r  = p - b * DHW;
    unsigned hw = r % HW;                 // depth index unused
    unsigned h  = hw / WWID;
    unsigned w  = hw - h * WWID;

    // semantic_mask[(b*N)][1][h][w] > 0.5  (scalar load + scalar branch)
    float sv = sem[b * (2 * HW) + HW + h * WWID + w];
    if (!(sv > 0.5f)) return;

    // g = (geom - (bx - dx/2)) / dx, truncate toward zero (== astype int32)
    float gxf = (geom[3u * p + 0] - (bx[0] - dx[0] * 0.5f)) / dx[0];
    float gyf = (geom[3u * p + 1] - (bx[1] - dx[1] * 0.5f)) / dx[1];
    float gzf = (geom[3u * p + 2] - (bx[2] - dx[2] * 0.5f)) / dx[2];
    int ix = (int)gxf, iy = (int)gyf, iz = (int)gzf;
    if (ix < 0 || ix >= NXG || iy < 0 || iy >= NYG || iz != 0) return; // NZ==1

    // Single-use feature stream: non-temporal so it cannot evict the
    // L2-resident accumulator.
    const f2 v = __builtin_nontemporal_load(
        (const f2*)(x + (size_t)p * CC + 2u * lane));
    float* dst = grid + ((size_t)((b * NXG + (unsigned)ix) * NYG + (unsigned)iy)) * CC
                      + 2u * lane;
    fadd_native(dst,     v.x);
    fadd_native(dst + 1, v.y);
}

// ---------------------------------------------------------------------------
// Fallback scatter (channel-major, straight into d_out) if ws too small.
// ---------------------------------------------------------------------------
__global__ void bev_scatter_chan(const float* __restrict__ x,
                                 const float* __restrict__ geom,
                                 const float* __restrict__ sem,
                                 const float* __restrict__ dx,
                                 const float* __restrict__ bx,
                                 float* __restrict__ out) {
    unsigned tid  = blockIdx.x * blockDim.x + threadIdx.x;
    unsigned lane = tid & 31u;
    unsigned p = (unsigned)__builtin_amdgcn_readfirstlane((int)(tid >> 5));
    if (p >= NPRIME) return;

    unsigned b  = p / DHW;
    unsigned r  = p - b * DHW;
    unsigned hw = r % HW;
    unsigned h  = hw / WWID;
    unsigned w  = hw - h * WWID;

    float sv = sem[b * (2 * HW) + HW + h * WWID + w];
    if (!(sv > 0.5f)) return;

    float gxf = (geom[3u * p + 0] - (bx[0] - dx[0] * 0.5f)) / dx[0];
    float gyf = (geom[3u * p + 1] - (bx[1] - dx[1] * 0.5f)) / dx[1];
    float gzf = (geom[3u * p + 2] - (bx[2] - dx[2] * 0.5f)) / dx[2];
    int ix = (int)gxf, iy = (int)gyf, iz = (int)gzf;
    if (ix < 0 || ix >= NXG || iy < 0 || iy >= NYG || iz != 0) return;

    const f2 v = __builtin_nontemporal_load(
        (const f2*)(x + (size_t)p * CC + 2u * lane));
    float* base = out + ((size_t)(b * CC + 2u * lane)) * PLANE
                      + (unsigned)ix * NYG + (unsigned)iy;
    fadd_native(base,         v.x);
    fadd_native(base + PLANE, v.y);
}

// ---------------------------------------------------------------------------
// Transpose (b, xy, c) -> (b, c, xy) in 64x64 f32 tiles.
// Stage 1: async global->LDS copies (global_load_async_to_lds_b128, ASYNCcnt;
//          512B contiguous global reads per wave per op; the async engine
//          owns the LDS writes). LDS rows padded to 68 dwords (16B aligned).
// Stage 2: wave-per-channel LDS reads (<=2-way bank conflict), 128B-contiguous
//          non-temporal global stores (output is write-once).
// ---------------------------------------------------------------------------
__global__ void bev_transpose_tile(const float* __restrict__ grid,
                                   float* __restrict__ out) {
    __shared__ float tile[64 * 68];      // 17,408 B, row stride 68 dwords
    const unsigned t   = threadIdx.x;    // 256 threads = 8 waves
    const unsigned b   = blockIdx.x / (PLANE / 64);
    const unsigned xy0 = (blockIdx.x % (PLANE / 64)) * 64;
    const float* src = grid + ((size_t)b * PLANE + xy0) * CC;

    // Stage 1: 1024 float4 elements / 256 threads = 4 async B128 per thread.
#pragma unroll
    for (int k = 0; k < 4; ++k) {
        unsigned e4 = (unsigned)k * 256u + t;   // 0..1023
        unsigned xy = e4 >> 4;                  // 0..63
        unsigned c4 = (e4 & 15u) << 2;          // 0,4,...,60
        const float* g = src + xy * CC + c4;    // coalesced across the wave
        unsigned lds = (unsigned)(uintptr_t)(&tile[xy * 68u + c4]); // LDS offset
        asm volatile("global_load_async_to_lds_b128 %0, %1, off"
                     :: "v"(lds), "v"(g)
                     : "memory");
    }
    asm volatile("s_wait_asynccnt 0" ::: "memory");
    __syncthreads();

    // Stage 2: wave wv handles channels wv*8 .. wv*8+7.
    const unsigned lane = t & 31u;
    const unsigned wv   = t >> 5;
    float* obase = out + ((size_t)b * CC) * PLANE + xy0;
#pragma unroll
    for (int j = 0; j < 8; ++j) {
        unsigned ch = wv * 8u + (unsigned)j;
        float v0 = tile[lane * 68u + ch];
        float v1 = tile[(lane + 32u) * 68u + ch];
        float* o = obase + (size_t)ch * PLANE;
        __builtin_nontemporal_store(v0, o + lane);        // 128B/wave/op
        __builtin_nontemporal_store(v1, o + lane + 32u);
    }
}

// ---------------------------------------------------------------------------
extern "C" void kernel_launch(void* const* d_in, const int* in_sizes, int n_in,
                              void* d_out, int out_size, void* d_ws, size_t ws_size,
                              hipStream_t stream) {
    const float* x    = (const float*)d_in[0];
    const float* geom = (const float*)d_in[1];
    const float* sem  = (const float*)d_in[2];
    const float* dx   = (const float*)d_in[3];
    const float* bx   = (const float*)d_in[4];
    float* out = (float*)d_out;

    const int zero_blocks    = (GRID_ELEMS / 4) / 256;            // 10,000
    const int scatter_blocks = (NPRIME * 32) / 256;               // 67,584
    const int trans_blocks   = BB * (PLANE / 64);                 // 2,500
    const size_t need = (size_t)GRID_ELEMS * sizeof(float);

    if (ws_size >= need) {
        float* grid = (float*)d_ws;
        bev_zero_f4<<<zero_blocks, 256, 0, stream>>>((f4*)grid, GRID_ELEMS / 4);
        bev_scatter_vox<<<scatter_blocks, 256, 0, stream>>>(x, geom, sem, dx, bx, grid);
        bev_transpose_tile<<<trans_blocks, 256, 0, stream>>>(grid, out);
    } else {
        bev_zero_f4<<<zero_blocks, 256, 0, stream>>>((f4*)out, GRID_ELEMS / 4);
        bev_scatter_chan<<<scatter_blocks, 256, 0, stream>>>(x, geom, sem, dx, bx, out);
    }
}